// PredRegress_29970281791566
// MI455X (gfx1250) — compile-verified
//
#include <hip/hip_runtime.h>
#include <math.h>

typedef __attribute__((ext_vector_type(2))) float v2f;
typedef __attribute__((ext_vector_type(8))) float v8f;

// searchsorted(starts, key, side='right') - 1, where starts[i] = se[2*i]
__device__ __forceinline__ int csr_find(const int* __restrict__ se, int n, int key) {
  int lo = 0, hi = n;
  while (lo < hi) {
    int mid = (lo + hi) >> 1;
    if (se[2 * mid] <= key) lo = mid + 1; else hi = mid;
  }
  return lo - 1;
}

__global__ void zero_ws_kernel(float* __restrict__ ws, int n) {
  int i = blockIdx.x * blockDim.x + threadIdx.x;
  if (i < n) ws[i] = 0.0f;
}

// One wave handles 16 consecutive trajectories (one contiguous 16*8T-byte
// region of agent_futs_xy):
//   1) coalesced async copy global -> LDS (512 B per issue) into bank-padded
//      per-traj slots, tracked by ASYNCcnt
//   2) V_WMMA_F32_16X16X4_F32 time-reduction: A = 16x4 tile of per-step L2
//      distances (fed from LDS), B = all-ones => D[m,n] = sum_k A[m,k] + C
//   3) |output - ADE| + masked float-atomic segment sums
__global__ __launch_bounds__(256)
void traj_ade_kernel(const float* __restrict__ output,
                     const float* __restrict__ futs,
                     const float* __restrict__ gt,
                     const int* __restrict__ traj_se,
                     const int* __restrict__ cls_se,
                     const unsigned char* __restrict__ oracles,
                     const int* __restrict__ apply_mask_p,
                     float* __restrict__ num,
                     float* __restrict__ den,
                     int n_trajs, int n_lanes, int batch, int T) {
  extern __shared__ char sh[];  // waves_per_block * 16 * (8*T + 16) bytes

  const int lane    = threadIdx.x & 31;
  const int wv      = threadIdx.x >> 5;                 // wave index in block
  const int wave_id = (int)((blockIdx.x * blockDim.x + threadIdx.x) >> 5);
  const int base    = wave_id * 16;
  if (base >= n_trajs) return;  // wave-uniform guard (EXEC stays all-ones)

  const int bytesPerTraj  = 8 * T;                       // 640 for T=80
  const int pitch         = bytesPerTraj + 16;           // bank de-stagger pad
  const int chunksPerTraj = T / 2;                       // 16B chunks per traj
  char* slice = sh + (size_t)wv * (size_t)(16 * pitch);

  // ---- stage 16 trajs of futs into LDS with coalesced async copies ----
  {
    const size_t gbase = (size_t)base * (size_t)bytesPerTraj;
    const size_t gmax  = (size_t)n_trajs * (size_t)bytesPerTraj - 16;
    const int totalChunks = 16 * chunksPerTraj;          // 640 for T=80
    for (int c0 = 0; c0 < totalChunks; c0 += 32) {       // uniform trip count
      const int c      = c0 + lane;
      const int traj   = c / chunksPerTraj;
      const int within = (c - traj * chunksPerTraj) * 16;
      size_t gbyte = gbase + (size_t)c * 16;             // contiguous global
      if (gbyte > gmax) gbyte = gmax;                    // OOB clamp (tail)
      const unsigned long long gaddr =
          (unsigned long long)(uintptr_t)((const char*)futs + gbyte);
      const unsigned ldsAddr =
          (unsigned)(uintptr_t)(slice + traj * pitch + within);
      asm volatile("global_load_async_to_lds_b128 %0, %1, off"
                   :: "v"(ldsAddr), "v"(gaddr)
                   : "memory");
    }
    asm volatile("s_wait_asynccnt 0x0" ::: "memory");    // wave-private slice
  }

  // ---- per-lane ownership for the compute phase (traj = base + (lane&15)) ----
  int trajL = base + (lane & 15);
  if (trajL >= n_trajs) trajL = n_trajs - 1;
  const int laneL = csr_find(traj_se, n_lanes, trajL);   // lane_of_traj
  const int sampL = csr_find(cls_se, batch, laneL);      // sample_of_lane

  const int tOff = (lane < 16) ? 0 : 2;                  // A-matrix K split
  const char*  lp = slice + (lane & 15) * pitch;         // this lane's traj in LDS
  const float* gp = gt + (size_t)sampL * (size_t)(2 * T);

  v8f acc = {0.f, 0.f, 0.f, 0.f, 0.f, 0.f, 0.f, 0.f};
  const v2f bones = {1.0f, 1.0f};                        // B[k,n] = 1, any layout

  for (int t0 = 0; t0 < T; t0 += 4) {                    // uniform main loop
    const int t = t0 + tOff;
    const float4 f = *(const float4*)(lp + 8 * t);       // x_t,y_t,x_{t+1},y_{t+1}
    const float4 g = *(const float4*)(gp + 2 * t);
    const float dx0 = f.x - g.x, dy0 = f.y - g.y;
    const float dx1 = f.z - g.z, dy1 = f.w - g.w;
    v2f a;
    a.x = sqrtf(dx0 * dx0 + dy0 * dy0);                  // K = t   (t+2 hi half)
    a.y = sqrtf(dx1 * dx1 + dy1 * dy1);                  // K = t+1 (t+3 hi half)
    acc = __builtin_amdgcn_wmma_f32_16x16x4_f32(
        /*neg_a=*/false, a, /*neg_b=*/false, bones,
        /*c_mod=*/(short)0, acc, /*reuse_a=*/false, /*reuse_b=*/false);
  }

  // ---- extract per-traj sums from the C/D layout ----
  // Lanes 0-15 hold M=r in component r; lanes 16-31 hold M=8+r. Columns are
  // identical, so lane m (m<8) takes acc[m] (traj base+m) and lane 16+j takes
  // acc[j] (traj base+8+j).
  const int k = lane & 7;
  float s = acc[0];
  if (k == 1) s = acc[1];
  else if (k == 2) s = acc[2];
  else if (k == 3) s = acc[3];
  else if (k == 4) s = acc[4];
  else if (k == 5) s = acc[5];
  else if (k == 6) s = acc[6];
  else if (k == 7) s = acc[7];

  const int mpost = (lane < 16) ? lane : (lane - 8);
  const int trajP = base + mpost;
  const bool active = ((lane < 8) || (lane >= 16 && lane < 24)) && (trajP < n_trajs);

  if (active) {
    const int laneP = csr_find(traj_se, n_lanes, trajP);
    const int sampP = csr_find(cls_se, batch, laneP);
    const float ade   = s * (1.0f / (float)T);
    const float delta = fabsf(output[trajP] - ade);
    float mv = 1.0f;
    if (*apply_mask_p) mv = oracles[laneP] ? 1.0f : 0.0f;
    atomicAdd(&num[sampP], delta * mv);
    atomicAdd(&den[sampP], mv);
  }
}

__global__ void finalize_kernel(const float* __restrict__ num,
                                const float* __restrict__ den,
                                float* __restrict__ out, int batch) {
  __shared__ float shr[1024];
  const int i = threadIdx.x;
  float v = 0.0f;
  if (i < batch) v = num[i] / den[i];
  shr[i] = v;
  __syncthreads();
  for (int stride = blockDim.x >> 1; stride > 0; stride >>= 1) {
    if (i < stride) shr[i] += shr[i + stride];
    __syncthreads();
  }
  if (i == 0) out[0] = shr[0] / (float)batch;
}

extern "C" void kernel_launch(void* const* d_in, const int* in_sizes, int n_in,
                              void* d_out, int out_size, void* d_ws, size_t ws_size,
                              hipStream_t stream) {
  const float*         output  = (const float*)d_in[0];          // [N_TRAJS, 1]
  const float*         futs    = (const float*)d_in[1];          // [N_TRAJS, T, 2]
  const float*         gt      = (const float*)d_in[2];          // [B, T, 2]
  const int*           traj_se = (const int*)d_in[3];            // [N_LANES, 2]
  const int*           cls_se  = (const int*)d_in[4];            // [B, 2]
  const unsigned char* oracles = (const unsigned char*)d_in[5];  // [N_LANES] bool
  const int*           apply   = (const int*)d_in[6];            // scalar

  const int n_trajs = in_sizes[0];
  const int n_lanes = in_sizes[3] / 2;
  const int batch   = in_sizes[4] / 2;
  const int T       = in_sizes[1] / (n_trajs * 2);

  float* num = (float*)d_ws;
  float* den = num + batch;

  // 1) zero the accumulators (ws is poisoned, not re-zeroed between replays)
  {
    const int n = 2 * batch;
    zero_ws_kernel<<<(n + 255) / 256, 256, 0, stream>>>(num, n);
  }

  // 2) main kernel: one wave32 per 16 trajectories, async-staged through LDS
  {
    const int waves   = (n_trajs + 15) / 16;
    const long long threads = (long long)waves * 32;
    const int block   = 256;
    const int grid    = (int)((threads + block - 1) / block);
    const int wavesPerBlock = block / 32;
    const size_t shbytes = (size_t)wavesPerBlock * 16 * (size_t)(8 * T + 16);
    traj_ade_kernel<<<grid, block, shbytes, stream>>>(output, futs, gt, traj_se,
                                                      cls_se, oracles, apply,
                                                      num, den,
                                                      n_trajs, n_lanes, batch, T);
  }

  // 3) mean over samples
  {
    int bdim = 1;
    while (bdim < batch && bdim < 1024) bdim <<= 1;
    finalize_kernel<<<1, bdim, 0, stream>>>(num, den, (float*)d_out, batch);
  }
}